// MLDRLoss_80187039416586
// MI455X (gfx1250) — compile-verified
//
#include <hip/hip_runtime.h>
#include <cmath>

typedef __attribute__((ext_vector_type(2))) float v2f;
typedef __attribute__((ext_vector_type(8))) float v8f;

#define T_LEN   (1u << 20)
#define NROW    8
#define BLK     256
#define NBLK    4096            // T_LEN / BLK
#define NCOEF   6
#define NCHAIN  (NCOEF * 2 * NROW)   // 96
#define EPS_X   1e-8f

struct Coefs {
    float c[6];      // coefficient (1 - a)
    float a[6];      // a = exp(-2200/(tau*sr))
    float a16[6];    // a^16
    float a256[6];   // a^256
    float a32k[6];   // a^32768 (0.0f when it underflows -- exact enough in f32)
    float l2a[6];    // log2(a)
};

// lane-constant broadcast read: lowers to v_readlane_b32 (no LDS path)
__device__ __forceinline__ float lane_read(float v, int lane) {
    return __uint_as_float(__builtin_amdgcn_readlane(__float_as_uint(v), lane));
}

// x[t] = clip( 0.5 * (ch0 +/- ch1)^2 , EPS ) for two consecutive t (t even)
__device__ __forceinline__ v2f xval2(const float* __restrict__ in, int bt, int m, int t) {
    const float2 c0 = *(const float2*)(in + (((size_t)(bt * 2 + 0)) << 20) + t);
    const float2 c1 = *(const float2*)(in + (((size_t)(bt * 2 + 1)) << 20) + t);
    v2f h, x;
    if (m) { h.x = c0.x - c1.x; h.y = c0.y - c1.y; }
    else   { h.x = c0.x + c1.x; h.y = c0.y + c1.y; }
    x.x = fmaxf(0.5f * h.x * h.x, EPS_X);
    x.y = fmaxf(0.5f * h.y * h.y, EPS_X);
    return x;
}

__device__ __forceinline__ float waveSum(float v) {
    #pragma unroll
    for (int off = 16; off > 0; off >>= 1) v += __shfl_down(v, off, 32);
    return v;
}

// branchless c * a^t (t in [-16,15]; negative -> 0) via repeated-squaring selects
__device__ __forceinline__ float apow_sel(int t, float c, float a, float a2,
                                          float a4, float a8) {
    float p = c;
    p *= (t & 1) ? a  : 1.0f;
    p *= (t & 2) ? a2 : 1.0f;
    p *= (t & 4) ? a4 : 1.0f;
    p *= (t & 8) ? a8 : 1.0f;
    return (t < 0) ? 0.0f : p;
}

// ---------------- K1: block-end partials for all 6 coefs -------------------
__global__ void k_block_partials(const float* __restrict__ pred,
                                 const float* __restrict__ tru,
                                 float* __restrict__ e, Coefs cf) {
    const int wg  = blockIdx.x;
    const int blk = wg & (NBLK - 1);
    const int sr  = wg >> 12;            // NBLK = 2^12
    const int row = sr & 7;
    const int sig = sr >> 3;
    const float* in = sig ? tru : pred;
    const int bt = row >> 1, m = row & 1;
    const int l  = threadIdx.x;
    const int t0 = blk * BLK + l * 8;

    float x[8];
    #pragma unroll
    for (int d = 0; d < 8; d += 2) {
        v2f p = xval2(in, bt, m, t0 + d);
        x[d] = p.x; x[d + 1] = p.y;
    }
    #pragma unroll
    for (int c = 0; c < 6; ++c) {
        const float a = cf.a[c];
        float acc = 0.f;                              // Horner: sum x[d]*a^(7-d)
        #pragma unroll
        for (int d = 0; d < 8; ++d) acc = acc * a + x[d];
        const float w = exp2f(cf.l2a[c] * (float)(248 - 8 * l));  // a^(248-8l)
        const float tot = waveSum(acc * w);
        if (l == 0) {
            const int chain = (c * 2 + sig) * NROW + row;
            e[(size_t)chain * NBLK + blk] = cf.c[c] * tot;
        }
    }
}

// ------- K2: wave-parallel carry scan (128 serial / 32-lane scan / 128) ----
__global__ void k_carry_scan(const float* __restrict__ e, float* __restrict__ cin,
                             Coefs cf) {
    const int chain = blockIdx.x;                     // one wave per chain
    const int lane  = threadIdx.x;
    const int cidx  = chain >> 4;                     // chain = coef*16 + sig*8 + row
    const float a256 = cf.a256[cidx];
    const float A128 = cf.a32k[cidx];                 // a256^128
    const float* ec = e  + (size_t)chain * NBLK + lane * 128;
    float*       cc = cin + (size_t)chain * NBLK + lane * 128;

    float yl = 0.f;                                   // local segment end (zero init)
    #pragma unroll 8
    for (int d = 0; d < 128; ++d) yl = ec[d] + a256 * yl;

    float yprev = 0.f, my_cin = 0.f;                  // exclusive cross-lane scan
    #pragma unroll
    for (int u = 0; u < 32; ++u) {
        const float eu = lane_read(yl, u);
        if (u == lane) my_cin = yprev;
        yprev = eu + A128 * yprev;
    }

    float carry = my_cin;                             // replay with true carry
    #pragma unroll 8
    for (int d = 0; d < 128; ++d) {
        cc[d] = carry;
        carry = ec[d] + a256 * carry;
    }
}

// ---- exact 256-sample EMA via 4x V_WMMA_F32_16X16X4_F32 + carry fixup -----
// Yl = M @ X, M[m][j] = c*a^(m-j) (lower tri 16x16), X[j][n] = x[t0+16n+j].
// Lane l: g=l>>4, n=l&15; D vgpr v holds Yl[v+8g][n]. Then add segment/block
// carry: y[k][n] = Yl[k][n] + carry_in(n) * a^(k+1).
__device__ __forceinline__ void ema_block_wmma(const float* __restrict__ in,
                                               int bt, int m, int t0,
                                               const v2f Aop[4],
                                               float a, float a16, float wbase,
                                               float cin_blk, float lg[8]) {
    const int l = threadIdx.x;
    const int g = l >> 4;
    const int n = l & 15;
    v8f acc = {};
    #pragma unroll
    for (int c4 = 0; c4 < 4; ++c4) {
        const v2f B = xval2(in, bt, m, t0 + 16 * n + 4 * c4 + 2 * g);
        acc = __builtin_amdgcn_wmma_f32_16x16x4_f32(false, Aop[c4], false, B,
                                                    (short)0, acc, false, false);
    }
    // segment-end values live in lanes 16..31, element 7 (k = 15)
    const float e7 = acc[7];
    float yprev = cin_blk, my_cin = 0.f;
    #pragma unroll
    for (int u = 0; u < 16; ++u) {
        const float eu = lane_read(e7, 16 + u);       // v_readlane_b32
        if (u == n) my_cin = yprev;
        yprev = eu + a16 * yprev;
    }
    float w = wbase;                                  // a^(8g+1)
    #pragma unroll
    for (int v = 0; v < 8; ++v) {
        lg[v] = __logf(acc[v] + my_cin * w);
        w *= a;
    }
}

// branchless, once-per-kernel A-operand build (shared by both signals)
__device__ __forceinline__ void build_aops(int n, int g, float c, float a,
                                           v2f Aop[4], float* wbase) {
    const float a2 = a * a, a4 = a2 * a2, a8 = a4 * a4;
    #pragma unroll
    for (int c4 = 0; c4 < 4; ++c4) {
        const int j0 = 4 * c4 + 2 * g;
        Aop[c4].x = apow_sel(n - j0,     c, a, a2, a4, a8);
        Aop[c4].y = apow_sel(n - j0 - 1, c, a, a2, a4, a8);
    }
    *wbase = g ? a * a8 : a;                          // a^(8g+1)
}

// ---------------- K3: long EMA -> D_l = log(yp) - log(yt) ------------------
__global__ void k_long_dl(const float* __restrict__ pred, const float* __restrict__ tru,
                          const float* __restrict__ cin, float* __restrict__ DL,
                          Coefs cf, int cidx) {
    const int wg  = blockIdx.x;
    const int blk = wg & (NBLK - 1);
    const int row = wg >> 12;
    const int bt = row >> 1, m = row & 1;
    const int t0 = blk * BLK;
    const float c = cf.c[cidx], a = cf.a[cidx], a16 = cf.a16[cidx];
    const float cin_p = cin[(size_t)((cidx * 2 + 0) * NROW + row) * NBLK + blk];
    const float cin_t = cin[(size_t)((cidx * 2 + 1) * NROW + row) * NBLK + blk];
    const int l = threadIdx.x, g = l >> 4, n = l & 15;
    v2f Aop[4]; float wbase;
    build_aops(n, g, c, a, Aop, &wbase);
    float lp[8], lt[8];
    ema_block_wmma(pred, bt, m, t0, Aop, a, a16, wbase, cin_p, lp);
    ema_block_wmma(tru,  bt, m, t0, Aop, a, a16, wbase, cin_t, lt);
    float* dst = DL + (size_t)row * T_LEN + t0 + 16 * n + 8 * g;
    *(float4*)(dst)     = make_float4(lp[0]-lt[0], lp[1]-lt[1], lp[2]-lt[2], lp[3]-lt[3]);
    *(float4*)(dst + 4) = make_float4(lp[4]-lt[4], lp[5]-lt[5], lp[6]-lt[6], lp[7]-lt[7]);
}

// ---------------- K4: short EMA + |D_s - D_l(shift)| partial sums ----------
__global__ void k_short_reduce(const float* __restrict__ pred, const float* __restrict__ tru,
                               const float* __restrict__ cin, const float* __restrict__ DL,
                               float* __restrict__ partials, Coefs cf,
                               int cidx, int shift) {
    const int wg  = blockIdx.x;
    const int blk = wg & (NBLK - 1);
    const int row = wg >> 12;
    const int bt = row >> 1, m = row & 1;
    const int t0 = blk * BLK;
    const float c = cf.c[cidx], a = cf.a[cidx], a16 = cf.a16[cidx];
    const float cin_p = cin[(size_t)((cidx * 2 + 0) * NROW + row) * NBLK + blk];
    const float cin_t = cin[(size_t)((cidx * 2 + 1) * NROW + row) * NBLK + blk];
    const int l = threadIdx.x, g = l >> 4, n = l & 15;
    v2f Aop[4]; float wbase;
    build_aops(n, g, c, a, Aop, &wbase);
    float lp[8], lt[8];
    ema_block_wmma(pred, bt, m, t0, Aop, a, a16, wbase, cin_p, lp);
    ema_block_wmma(tru,  bt, m, t0, Aop, a, a16, wbase, cin_t, lt);
    const int tbase = t0 + 16 * n + 8 * g;
    const float* dlr = DL + (size_t)row * T_LEN;
    float s = 0.f;
    #pragma unroll
    for (int v = 0; v < 8; ++v) {
        int j = tbase + v + shift;
        if (j >= (int)T_LEN) j -= (int)T_LEN;
        s += fabsf((lp[v] - lt[v]) - dlr[j]);
    }
    const float tot = waveSum(s);
    if (threadIdx.x == 0) partials[(size_t)row * NBLK + blk] = tot;
}

// ---------------- K5: deterministic final reduction ------------------------
__global__ void k_final(const float* __restrict__ partials, float* __restrict__ out) {
    __shared__ float sm[256];
    const int tid = threadIdx.x;
    const int total = 3 * NROW * NBLK;
    float s = 0.f;
    for (int i = tid; i < total; i += 256) s += partials[i];
    sm[tid] = s;
    __syncthreads();
    #pragma unroll
    for (int off = 128; off > 0; off >>= 1) {
        if (tid < off) sm[tid] += sm[tid + off];
        __syncthreads();
    }
    if (tid == 0) out[0] = sm[0] / 8388608.0f;   // mean over [8, 2^20] per pair
}

extern "C" void kernel_launch(void* const* d_in, const int* in_sizes, int n_in,
                              void* d_out, int out_size, void* d_ws, size_t ws_size,
                              hipStream_t stream) {
    const float* pred = (const float*)d_in[0];
    const float* tru  = (const float*)d_in[1];
    float* out = (float*)d_out;
    float* ws  = (float*)d_ws;

    float* e        = ws;
    float* cin      = e   + (size_t)NCHAIN * NBLK;
    float* DL       = cin + (size_t)NCHAIN * NBLK;
    float* partials = DL  + (size_t)NROW * T_LEN;

    const double SR = 44100.0;
    const double taus[6] = {10.0, 500.0, 50.0, 1500.0, 100.0, 3000.0}; // [2p]=short,[2p+1]=long
    Coefs cf;
    for (int i = 0; i < 6; ++i) {
        const float cfl = (float)(1.0 - std::exp(-2200.0 / (taus[i] * SR)));
        const float afl = 1.0f - cfl;
        cf.c[i]    = cfl;
        cf.a[i]    = afl;
        cf.a16[i]  = (float)std::pow((double)afl, 16.0);
        cf.a256[i] = (float)std::pow((double)afl, 256.0);
        cf.a32k[i] = (float)std::pow((double)afl, 32768.0);
        cf.l2a[i]  = (float)(std::log((double)afl) / std::log(2.0));
    }
    const double sms[3] = {10.0, 50.0, 100.0}, lms[3] = {500.0, 1500.0, 3000.0};
    int shifts[3];
    for (int p = 0; p < 3; ++p) shifts[p] = (int)(SR * (lms[p] - sms[p]) * 0.0005);

    k_block_partials<<<2 * NROW * NBLK, 32, 0, stream>>>(pred, tru, e, cf);
    k_carry_scan<<<NCHAIN, 32, 0, stream>>>(e, cin, cf);
    for (int p = 0; p < 3; ++p) {
        k_long_dl<<<NROW * NBLK, 32, 0, stream>>>(pred, tru, cin, DL, cf, 2 * p + 1);
        k_short_reduce<<<NROW * NBLK, 32, 0, stream>>>(pred, tru, cin, DL,
                partials + (size_t)p * NROW * NBLK, cf, 2 * p, shifts[p]);
    }
    k_final<<<1, 256, 0, stream>>>(partials, out);
}